// InfluenceGraphConv_52828097741226
// MI455X (gfx1250) — compile-verified
//
#include <hip/hip_runtime.h>

typedef __attribute__((ext_vector_type(2))) float v2f;
typedef __attribute__((ext_vector_type(8))) float v8f;

#define D_IN  128
#define D_OUT 128

// ---------------------------------------------------------------------------
// Kernel 1: h[row, :] = (feat[row, :] @ W) * cu[row]
// Block: 256 threads (8 waves). Each wave computes one 16x16 tile of h using
// V_WMMA_F32_16X16X4_F32 chained over K=128 (32 WMMA ops). W staged in LDS.
// ---------------------------------------------------------------------------
__global__ __launch_bounds__(256) void influence_gemm_cu_kernel(
    const float* __restrict__ feat, const float* __restrict__ W,
    const float* __restrict__ cu, float* __restrict__ h, int nrows) {
  __shared__ float sW[D_IN * D_OUT];  // 64 KB: W row-major [k][n]

  const int tid = threadIdx.x;
  // Cooperative, coalesced load of W into LDS: 16384 floats / 256 threads.
#pragma unroll
  for (int i = 0; i < (D_IN * D_OUT) / 256; ++i)
    sW[tid + i * 256] = W[tid + i * 256];
  __syncthreads();

  const int wave    = tid >> 5;
  const int lane    = tid & 31;
  const int rowBase = blockIdx.x * 16;
  const int colBase = wave * 16;      // 8 waves * 16 = 128 = D_OUT
  const int m       = lane & 15;      // M (A) / N (B,C) index within tile
  const int kh      = (lane >> 4) * 2;  // lanes 16-31 hold K-pair {2,3}

  // Clamp A-row for tail safety (keeps EXEC all-ones across WMMA).
  int arow_idx = rowBase + m;
  if (arow_idx >= nrows) arow_idx = nrows - 1;
  const float* arow = feat + (size_t)arow_idx * D_IN + kh;

  v8f acc = {};
#pragma unroll
  for (int kk = 0; kk < D_IN / 4; ++kk) {
    // A fragment: 16x4 f32. VGPR0 = K {0|2}, VGPR1 = K {1|3} (per lane half).
    v2f A = *(const v2f*)(arow + kk * 4);
    // B fragment: 4x16 f32. Row K striped across lanes within a VGPR.
    const int kb = kk * 4 + kh;
    v2f B;
    B.x = sW[kb * D_OUT + colBase + m];
    B.y = sW[(kb + 1) * D_OUT + colBase + m];
    acc = __builtin_amdgcn_wmma_f32_16x16x4_f32(
        /*neg_a=*/false, A, /*neg_b=*/false, B,
        /*c_mod=*/(short)0, acc, /*reuse_a=*/false, /*reuse_b=*/false);
  }

  // C/D layout: lanes 0-15 hold rows M=0..7 in acc[0..7]; lanes 16-31 rows 8..15.
  const int mBase = rowBase + (lane >> 4) * 8;
#pragma unroll
  for (int v = 0; v < 8; ++v) {
    const int row = mBase + v;
    if (row < nrows)
      h[(size_t)row * D_OUT + colBase + m] = acc[v] * cu[row];
  }
}

// ---------------------------------------------------------------------------
// Kernel 2: for each edge e: out[dst[e], :] += h[src[e], :] * edge_w[e]*cv[dst]
// One wave (32 lanes) per edge; each lane owns a float4 chunk (32*4 = 128).
// cv fused into the edge weight (distributes over the segment sum).
// ---------------------------------------------------------------------------
__global__ __launch_bounds__(256) void influence_scatter_kernel(
    const float* __restrict__ h, const int* __restrict__ src,
    const int* __restrict__ dst, const float* __restrict__ edge_w,
    const float* __restrict__ cv, float* __restrict__ out, int nedges) {
  const long gid = (long)blockIdx.x * blockDim.x + threadIdx.x;
  const long e = gid >> 5;
  if (e >= nedges) return;
  const int lane = (int)(gid & 31);

  const int s = src[e];
  const int d = dst[e];
  const float w = edge_w[e] * cv[d];

  const float4 hv = *(const float4*)(h + (size_t)s * D_OUT + lane * 4);
  float* op = out + (size_t)d * D_OUT + lane * 4;
  // Non-returning global_atomic_add_f32 x4 (tracked on STOREcnt).
  atomicAdd(op + 0, hv.x * w);
  atomicAdd(op + 1, hv.y * w);
  atomicAdd(op + 2, hv.z * w);
  atomicAdd(op + 3, hv.w * w);
}

extern "C" void kernel_launch(void* const* d_in, const int* in_sizes, int n_in,
                              void* d_out, int out_size, void* d_ws, size_t ws_size,
                              hipStream_t stream) {
  const float* feat = (const float*)d_in[0];
  const float* W    = (const float*)d_in[1];
  const float* cu   = (const float*)d_in[2];
  const float* cv   = (const float*)d_in[3];
  const float* ew   = (const float*)d_in[4];
  const int*   src  = (const int*)d_in[5];
  const int*   dst  = (const int*)d_in[6];

  float* out = (float*)d_out;
  float* h   = (float*)d_ws;  // [N, 128] f32 scratch: N*128*4 = 51.2 MB

  const int N = in_sizes[2];  // cu has N elements
  const int E = in_sizes[5];  // src has E elements

  // Zero the accumulator output (harness poisons d_out). Capturable memset.
  hipMemsetAsync(d_out, 0, (size_t)out_size * sizeof(float), stream);

  const int rowTiles = (N + 15) / 16;
  influence_gemm_cu_kernel<<<rowTiles, 256, 0, stream>>>(feat, W, cu, h, N);

  const long threads = (long)E * 32;
  const int tpb = 256;
  const long blocks = (threads + tpb - 1) / tpb;
  influence_scatter_kernel<<<(int)blocks, tpb, 0, stream>>>(h, src, dst, ew,
                                                            cv, out, E);
}